// Sequence_19748259627556
// MI455X (gfx1250) — compile-verified
//
#include <hip/hip_runtime.h>
#include <math.h>

// ---------------- problem constants (from reference) ----------------
#define Hh    51          // hidden size
#define HP    52          // padded K (multiple of 4 for 16x16x4 f32 WMMA)
#define G4    204         // 4*H gate width
#define G4P   208         // padded gate width (13 tiles of 16)
#define NTILE 13
#define Bsz   4096
#define Tin   1024
#define ROWS  16          // batch rows per block (WMMA M)
#define NW    4           // waves per workgroup (split the N-tiles)
#define NTHR  (32 * NW)
#define JIT   7           // max hidden-unit iterations per thread (ceil(51/8))
#define XCH   32          // x staging chunk in timesteps (divides Tin)

typedef float v2f __attribute__((ext_vector_type(2)));
typedef float v4f __attribute__((ext_vector_type(4)));
typedef float v8f __attribute__((ext_vector_type(8)));

// hardware tanh on CDNA5 (v_tanh_f32); guarded fallback keeps compile safe
#if __has_builtin(__builtin_amdgcn_tanhf)
__device__ __forceinline__ float htanh(float v) { return __builtin_amdgcn_tanhf(v); }
#elif __has_builtin(__builtin_amdgcn_tanh_f32)
__device__ __forceinline__ float htanh(float v) { return __builtin_amdgcn_tanh_f32(v); }
#else
__device__ __forceinline__ float htanh(float v) { return tanhf(v); }
#endif
// sigmoid(x) = 0.5*tanh(x/2) + 0.5  -> one TRANS op instead of exp+rcp
__device__ __forceinline__ float sigm(float v) { return fmaf(0.5f, htanh(0.5f * v), 0.5f); }

// D = A(16x4 f32) * B(4x16 f32) + C, full f32 precision on the matrix core
__device__ __forceinline__ v8f wmma4(v2f a, v2f b, v8f c) {
  return __builtin_amdgcn_wmma_f32_16x16x4_f32(false, a, false, b, (short)0, c,
                                               false, false);
}

struct Smem {
  float Whh1[G4P * HP];    // padded row-major [n][k], zeros in pad
  float Wih2[G4P * HP];
  float Whh2[G4P * HP];
  float wih1[G4P];         // W_ih1[:,0]
  float b1[G4P];           // b_ih1 + b_hh1 (folded into WMMA C operand)
  float b2[G4P];           // b_ih2 + b_hh2
  float wlin[HP];
  float blin;
  float h1[ROWS * HP];     // hidden states, row-major [m][k], pad col = 0
  float h2[ROWS * HP];     // (cell states c1/c2 live in registers)
  float gates[G4P * ROWS]; // COLUMN-major [N][M]: WMMA tile -> 2x ds_store_b128
  float xstage[ROWS * XCH];// staged x chunk: [m][t&31]
  float psum[NTHR];        // per-thread partials of the output projection
  float xb[ROWS];          // fed-back output per row (future phase input)
};

__global__ __launch_bounds__(NTHR, 1)
void lstm_seq_kernel(const float* __restrict__ x,
                     const float* __restrict__ W_ih1,
                     const float* __restrict__ W_hh1,
                     const float* __restrict__ b_ih1,
                     const float* __restrict__ b_hh1,
                     const float* __restrict__ W_ih2,
                     const float* __restrict__ W_hh2,
                     const float* __restrict__ b_ih2,
                     const float* __restrict__ b_hh2,
                     const float* __restrict__ W_lin,
                     const float* __restrict__ b_lin,
                     float* __restrict__ out,
                     int t_out) {
  __shared__ Smem sm;
  const int tid  = threadIdx.x;
  const int lane = tid & 31;
  const int wid  = tid >> 5;
  const int half = lane >> 4;   // 0: lanes 0-15, 1: lanes 16-31
  const int lm   = lane & 15;
  const int em   = tid & 15;    // elementwise: batch row
  const int ej   = tid >> 4;    // elementwise: hidden-unit phase (0..7)

  // ---- cooperative setup: weights -> padded LDS, zero state ----
  for (int idx = tid; idx < G4P * HP; idx += NTHR) {
    int n = idx / HP, k = idx - n * HP;
    bool ok = (n < G4) && (k < Hh);
    sm.Whh1[idx] = ok ? W_hh1[n * Hh + k] : 0.0f;
    sm.Wih2[idx] = ok ? W_ih2[n * Hh + k] : 0.0f;
    sm.Whh2[idx] = ok ? W_hh2[n * Hh + k] : 0.0f;
  }
  for (int n = tid; n < G4P; n += NTHR) {
    bool ok = n < G4;
    sm.wih1[n] = ok ? W_ih1[n] : 0.0f;
    sm.b1[n]   = ok ? (b_ih1[n] + b_hh1[n]) : 0.0f;
    sm.b2[n]   = ok ? (b_ih2[n] + b_hh2[n]) : 0.0f;
  }
  for (int k = tid; k < HP; k += NTHR) sm.wlin[k] = (k < Hh) ? W_lin[k] : 0.0f;
  if (tid == 0) sm.blin = b_lin[0];
  for (int i = tid; i < 2 * ROWS * HP; i += NTHR) (&sm.h1[0])[i] = 0.0f;
  __syncthreads();

  // cell states in registers: thread <-> (em, j=ej+8*it) ownership is static
  float rc1[JIT], rc2[JIT];
  #pragma unroll
  for (int it = 0; it < JIT; ++it) { rc1[it] = 0.0f; rc2[it] = 0.0f; }

  const int r0 = blockIdx.x * ROWS;

  for (int t = 0; t < t_out; ++t) {
    // stage 32 timesteps of x (contiguous per row) every 32 steps: 128 x b128
    if (t < Tin && (t & (XCH - 1)) == 0) {
      const int m  = tid >> 3;
      const int s4 = (tid & 7) * 4;
      v4f xv4 = *(const v4f*)&x[(size_t)(r0 + m) * Tin + t + s4];
      *(v4f*)&sm.xstage[m * XCH + s4] = xv4;
    }

    // ---------- layer 1 GEMM: gates = h1_prev @ W_hh1^T + b1 (waves split N-tiles) ----------
    #pragma unroll 1
    for (int nt = wid; nt < NTILE; nt += NW) {
      const int n0 = nt * 16;
      const float bv = sm.b1[n0 + lm];
      v8f acc = {bv, bv, bv, bv, bv, bv, bv, bv};
      #pragma unroll
      for (int kb = 0; kb < HP; kb += 4) {
        v2f a = *(const v2f*)&sm.h1[lm * HP + kb + 2 * half];
        v2f b = *(const v2f*)&sm.Whh1[(n0 + lm) * HP + kb + 2 * half];
        acc = wmma4(a, b, acc);
      }
      float* g = &sm.gates[(n0 + lm) * ROWS + 8 * half];
      *(v4f*)(g)     = __builtin_shufflevector(acc, acc, 0, 1, 2, 3);
      *(v4f*)(g + 4) = __builtin_shufflevector(acc, acc, 4, 5, 6, 7);
    }
    __syncthreads();   // A: gates1/xstage ready

    // ---------- elementwise layer 1 (i/f/g/o order; c1 in registers) ----------
    {
      const float xv = (t < Tin) ? sm.xstage[em * XCH + (t & (XCH - 1))] : sm.xb[em];
      #pragma unroll
      for (int it = 0; it < JIT; ++it) {
        const int j = ej + 8 * it;
        if (j < Hh) {
          float gi = sm.gates[(j         ) * ROWS + em] + xv * sm.wih1[j         ];
          float gf = sm.gates[(j +     Hh) * ROWS + em] + xv * sm.wih1[j +     Hh];
          float gg = sm.gates[(j + 2 * Hh) * ROWS + em] + xv * sm.wih1[j + 2 * Hh];
          float go = sm.gates[(j + 3 * Hh) * ROWS + em] + xv * sm.wih1[j + 3 * Hh];
          float c = sigm(gf) * rc1[it] + sigm(gi) * htanh(gg);
          rc1[it] = c;
          sm.h1[em * HP + j] = sigm(go) * htanh(c);
        }
      }
    }
    __syncthreads();   // B: h1 ready

    // ---------- layer 2 GEMM: gates = h1_new @ W_ih2^T + h2_prev @ W_hh2^T + b2 ----------
    #pragma unroll 1
    for (int nt = wid; nt < NTILE; nt += NW) {
      const int n0 = nt * 16;
      const float bv = sm.b2[n0 + lm];
      v8f acc = {bv, bv, bv, bv, bv, bv, bv, bv};
      #pragma unroll
      for (int kb = 0; kb < HP; kb += 4) {
        v2f a = *(const v2f*)&sm.h1[lm * HP + kb + 2 * half];
        v2f b = *(const v2f*)&sm.Wih2[(n0 + lm) * HP + kb + 2 * half];
        acc = wmma4(a, b, acc);
      }
      #pragma unroll
      for (int kb = 0; kb < HP; kb += 4) {
        v2f a = *(const v2f*)&sm.h2[lm * HP + kb + 2 * half];
        v2f b = *(const v2f*)&sm.Whh2[(n0 + lm) * HP + kb + 2 * half];
        acc = wmma4(a, b, acc);
      }
      float* g = &sm.gates[(n0 + lm) * ROWS + 8 * half];
      *(v4f*)(g)     = __builtin_shufflevector(acc, acc, 0, 1, 2, 3);
      *(v4f*)(g + 4) = __builtin_shufflevector(acc, acc, 4, 5, 6, 7);
    }
    __syncthreads();   // C: gates2 ready

    // ---------- elementwise layer 2 (c2 in regs) + fused output-projection partials ----------
    {
      float ps = 0.0f;
      #pragma unroll
      for (int it = 0; it < JIT; ++it) {
        const int j = ej + 8 * it;
        if (j < Hh) {
          float gi = sm.gates[(j         ) * ROWS + em];
          float gf = sm.gates[(j +     Hh) * ROWS + em];
          float gg = sm.gates[(j + 2 * Hh) * ROWS + em];
          float go = sm.gates[(j + 3 * Hh) * ROWS + em];
          float c = sigm(gf) * rc2[it] + sigm(gi) * htanh(gg);
          rc2[it] = c;
          float hh = sigm(go) * htanh(c);
          sm.h2[em * HP + j] = hh;
          ps = fmaf(hh, sm.wlin[j], ps);
        }
      }
      sm.psum[ej * ROWS + em] = ps;
    }
    __syncthreads();   // D: h2/psum ready, gates dead

    // ---------- output: 8-way reduce of partials + feedback ----------
    if (tid < ROWS) {
      float acc = sm.blin;
      #pragma unroll
      for (int e = 0; e < 8; ++e) acc += sm.psum[e * ROWS + tid];
      out[(size_t)(r0 + tid) * t_out + t] = acc;
      sm.xb[tid] = acc;     // fed back as input for t >= Tin (read after barrier A)
    }
  }
}

extern "C" void kernel_launch(void* const* d_in, const int* in_sizes, int n_in,
                              void* d_out, int out_size, void* d_ws, size_t ws_size,
                              hipStream_t stream) {
  (void)in_sizes; (void)n_in; (void)d_ws; (void)ws_size;
  const float* x     = (const float*)d_in[0];
  const float* W_ih1 = (const float*)d_in[1];
  const float* W_hh1 = (const float*)d_in[2];
  const float* b_ih1 = (const float*)d_in[3];
  const float* b_hh1 = (const float*)d_in[4];
  const float* W_ih2 = (const float*)d_in[5];
  const float* W_hh2 = (const float*)d_in[6];
  const float* b_ih2 = (const float*)d_in[7];
  const float* b_hh2 = (const float*)d_in[8];
  const float* W_lin = (const float*)d_in[9];
  const float* b_lin = (const float*)d_in[10];
  // future horizon derived deterministically from output size: out = [B, T+future]
  const int t_out = out_size / Bsz;

  dim3 grid(Bsz / ROWS);   // 256 workgroups, each: 4 waves cooperating on 16 rows
  dim3 block(NTHR);
  lstm_seq_kernel<<<grid, block, 0, stream>>>(x, W_ih1, W_hh1, b_ih1, b_hh1,
                                              W_ih2, W_hh2, b_ih2, b_hh2,
                                              W_lin, b_lin, (float*)d_out, t_out);
}